// GPUPatchify_38525856645238
// MI455X (gfx1250) — compile-verified
//
#include <hip/hip_runtime.h>
#include <stdint.h>

// ---------------------------------------------------------------------------
// GPUPatchify for MI455X (gfx1250, wave32).
//
// Phase 1 (bandwidth-bound, 201MB @ 23.3TB/s ~ 9us):
//   k_stats0: stream 16x16 tiles global->LDS via global_load_async_to_lds_b128,
//   then V_WMMA_F32_16X16X4_F32 computes 4-column-group sums Y = X*G
//   (G = kron(I4, ones(4,1)) zero-padded); 4x4 block sums of X and X^2 fall
//   out as per-lane adds of accumulator VGPRs (D-layout: rows->VGPR idx).
// Phase 2: moment pyramid (2x2 sums) + per-level SSE errors.
// Phase 3: deterministic fixed-pass greedy merge (bucket-threshold select).
// Phase 4: emit errors_flat (float) then lm (int32 bit-exact) into d_out.
// ---------------------------------------------------------------------------

typedef float v2f __attribute__((ext_vector_type(2)));
typedef float v8f __attribute__((ext_vector_type(8)));

#define CH     3
#define IMG_H  4096
#define IMG_W  4096
#define B0     4            // min_size
#define NB0    1024         // level-0 blocks per side
#define NLEV   11           // levels 0..10
#define NPASS  16           // merge passes (converges in ~5; extras no-op)
#define NBUCK  4096

__host__ __device__ constexpr int h_cellOff(int l) {
  int t = 0;
  for (int j = 0; j < l; ++j) { int n = NB0 >> j; t += n * n; }
  return t;
}
constexpr int TOTAL_CELLS = h_cellOff(NLEV);            // 1,398,101
constexpr int CAND_CELLS  = TOTAL_CELLS - NB0 * NB0;    //   349,525

__device__ __forceinline__ int d_cellOff(int l) {
  int t = 0;
  for (int j = 0; j < l; ++j) { int n = NB0 >> j; t += n * n; }
  return t;
}

// Locate level l (1..10) and in-level index for a flat candidate index.
__device__ __forceinline__ void d_locate(int idx, int& l, int& rem, int& n) {
  rem = idx; l = 1; n = NB0 >> 1;
  while (rem >= n * n) { rem -= n * n; ++l; n >>= 1; }
}

// ---------------------------------------------------------------------------
// Phase 1: per-4x4-block (S1, S2) per channel via async-LDS + WMMA.
// Grid: 3*256*256 tiles / 8 waves-per-block = 24576 blocks of 256 threads.
// ---------------------------------------------------------------------------
__global__ void k_stats0(const float* __restrict__ img,
                         float2* __restrict__ stats0) {
  __shared__ float tile[8][256];                 // 1KB tile per wave
  const int w    = threadIdx.x >> 5;
  const int lane = threadIdx.x & 31;
  const int t    = blockIdx.x * 8 + w;           // tile id, exact coverage
  const int c    = t >> 16;                      // channel (65536 tiles/ch)
  const int tt   = t & 65535;
  const int tr   = tt >> 8;                      // tile row 0..255
  const int tc   = tt & 255;                     // tile col 0..255

  const float* base = img + (size_t)c * IMG_H * IMG_W
                          + (size_t)tr * 16 * IMG_W + (size_t)tc * 16;

  // --- async stage: 64 x b128 chunks; 2 per lane (ASYNCcnt-tracked) ---
  const uint32_t ldsBase = (uint32_t)(size_t)&tile[w][0]; // low 32b = LDS off
  const int k0 = lane, k1 = lane + 32;
  uint64_t ga0 = (uint64_t)(size_t)(base + (k0 >> 2) * IMG_W + (k0 & 3) * 4);
  uint64_t ga1 = (uint64_t)(size_t)(base + (k1 >> 2) * IMG_W + (k1 & 3) * 4);
  uint32_t la0 = ldsBase + k0 * 16;
  uint32_t la1 = ldsBase + k1 * 16;
  asm volatile("global_load_async_to_lds_b128 %0, %1, off"
               :: "v"(la0), "v"(ga0) : "memory");
  asm volatile("global_load_async_to_lds_b128 %0, %1, off"
               :: "v"(la1), "v"(ga1) : "memory");
  asm volatile("s_wait_asynccnt 0" ::: "memory");   // per-wave counter

  // --- build fragments ---
  // A (16x4 f32) layout: lanes 0-15 hold M=lane, K={4i,4i+1}; lanes 16-31
  // hold M=lane-16, K={4i+2,4i+3}.
  const int arow     = lane & 15;
  const int acolBase = (lane >> 4) * 2;
  v2f a[4], asq[4], b[4];
#pragma unroll
  for (int i = 0; i < 4; ++i) {
    const float* p = &tile[w][arow * 16 + 4 * i + acolBase];
    float x0 = p[0], x1 = p[1];
    a[i][0]   = x0;      a[i][1]   = x1;
    asq[i][0] = x0 * x0; asq[i][1] = x1 * x1;
    // B chunk i = rows 4i..4i+3 of G, all identical: G[k][n] = (n==i).
    // n = lane%16 under any row->(vgpr,half) mapping -> layout-robust.
    float g = (arow == i) ? 1.0f : 0.0f;
    b[i][0] = g; b[i][1] = g;
  }

  // --- Y = X*G and Y2 = (X.^2)*G via 4-chunk K accumulation each ---
  v8f s1 = {0.f,0.f,0.f,0.f,0.f,0.f,0.f,0.f};
  v8f s2 = {0.f,0.f,0.f,0.f,0.f,0.f,0.f,0.f};
#pragma unroll
  for (int i = 0; i < 4; ++i) {
    s1 = __builtin_amdgcn_wmma_f32_16x16x4_f32(false, a[i],   false, b[i],
                                               (short)0, s1, false, false);
    s2 = __builtin_amdgcn_wmma_f32_16x16x4_f32(false, asq[i], false, b[i],
                                               (short)0, s2, false, false);
  }

  // D layout: VGPR v = row M (lanes0-15: M=v, lanes16-31: M=v+8), N=lane%16.
  // Summing VGPRs 0-3 / 4-7 collapses the 4-row groups per block row.
  float s1lo = s1[0] + s1[1] + s1[2] + s1[3];   // block rows 0 / 2
  float s1hi = s1[4] + s1[5] + s1[6] + s1[7];   // block rows 1 / 3
  float s2lo = s2[0] + s2[1] + s2[2] + s2[3];
  float s2hi = s2[4] + s2[5] + s2[6] + s2[7];

  if (arow < 4) {                                // N=0..3 hold block cols
    int bc = tc * 4 + arow;
    int br = tr * 4 + (lane >> 4) * 2;
    float2* o = stats0 + (size_t)c * NB0 * NB0;
    o[(size_t)br * NB0 + bc]       = make_float2(s1lo, s2lo);
    o[(size_t)(br + 1) * NB0 + bc] = make_float2(s1hi, s2hi);
  }
}

// ---------------------------------------------------------------------------
// Phase 2: moment pyramid + SSE errors
// ---------------------------------------------------------------------------
__global__ void k_agg(float2* stats, int l) {
  const int np = NB0 >> (l + 1), n = NB0 >> l;
  int idx = blockIdx.x * blockDim.x + threadIdx.x;
  if (idx >= np * np) return;
  int pr = idx / np, pc = idx % np;
  const float2* src = stats + (size_t)CH * d_cellOff(l);
  float2* dst       = stats + (size_t)CH * d_cellOff(l + 1);
  for (int c = 0; c < CH; ++c) {
    const float2* s = src + (size_t)c * n * n;
    float2 a00 = s[(size_t)(2*pr)   * n + 2*pc];
    float2 a01 = s[(size_t)(2*pr)   * n + 2*pc + 1];
    float2 a10 = s[(size_t)(2*pr+1) * n + 2*pc];
    float2 a11 = s[(size_t)(2*pr+1) * n + 2*pc + 1];
    dst[(size_t)c * np * np + idx] =
        make_float2(a00.x + a01.x + a10.x + a11.x,
                    a00.y + a01.y + a10.y + a11.y);
  }
}

__global__ void k_err(const float2* __restrict__ stats,
                      float* __restrict__ errors, int l) {
  const int n = NB0 >> l;
  int idx = blockIdx.x * blockDim.x + threadIdx.x;
  if (idx >= n * n) return;
  const float2* s = stats + (size_t)CH * d_cellOff(l);
  float bs  = (float)(B0 << l);
  float nn  = bs * bs;
  float fac = nn / (nn - 1.0f);        // unbiased var * n == SSE * n/(n-1)
  float e = 0.f;
  for (int c = 0; c < CH; ++c) {
    float2 v = s[(size_t)c * n * n + idx];
    e += v.y - v.x * v.x / nn;         // SSE = S2 - S1^2/n
  }
  errors[d_cellOff(l) + idx] = e * fac;
}

// ---------------------------------------------------------------------------
// Phase 3: deterministic batched greedy merge
// scal: 0=current 1=fixed 2=min_batch 3=batch_k 4=thrBucket 5=merges 6=done
// ---------------------------------------------------------------------------
__global__ void k_init(uint8_t* alive, int* scal, const int* fixedLen) {
  int idx = blockIdx.x * blockDim.x + threadIdx.x;
  if (idx < TOTAL_CELLS) alive[idx] = (idx < NB0 * NB0) ? 1 : 0;
  if (idx == 0) {
    int fx = fixedLen[0];
    scal[0] = NB0 * NB0; scal[1] = fx;
    int mb = (NB0 * NB0 - fx) / 20; if (mb < 64) mb = 64;
    scal[2] = mb; scal[3] = 0; scal[4] = 0; scal[5] = 0;
    scal[6] = (NB0 * NB0 <= fx) ? 1 : 0;
  }
}

__global__ void k_pass_setup(int* scal, int* hist) {
  for (int b = threadIdx.x; b < NBUCK; b += blockDim.x) hist[b] = 0;
  if (threadIdx.x == 0) {
    if (scal[0] <= scal[1]) scal[6] = 1;
    if (!scal[6]) {
      long long cur = scal[0], fx = scal[1], mb = scal[2];
      long long bk = (cur * 15) / 100;           // batch_ratio 0.15
      if (bk > 100000) bk = 100000;
      if (bk < mb) bk = mb;
      if (cur - fx < bk) { long long b2 = cur - fx + 100; bk = b2 > mb ? b2 : mb; }
      scal[3] = (int)bk;
    }
  }
}

__global__ void k_candidates(const float* __restrict__ errors,
                             const uint8_t* __restrict__ alive,
                             uint16_t* __restrict__ candB,
                             const int* __restrict__ scal, int* hist) {
  if (scal[6]) return;
  int idx = blockIdx.x * blockDim.x + threadIdx.x;
  if (idx >= CAND_CELLS) return;
  int l, rem, n; d_locate(idx, l, rem, n);
  int r = rem / n, c = rem % n, cn = n << 1;
  const uint8_t* ch = alive + d_cellOff(l - 1);
  int b = (2 * r) * cn + 2 * c;
  uint16_t bu = 0xFFFFu;
  if (ch[b] && ch[b + 1] && ch[b + cn] && ch[b + cn + 1]) {
    const float* ep = errors + d_cellOff(l);
    const float* ec = errors + d_cellOff(l - 1);
    float cost = ep[rem] - (ec[b] + ec[b+1] + ec[b+cn] + ec[b+cn+1]);
    uint32_t u = __float_as_uint(cost);            // order-preserving key
    u = (u & 0x80000000u) ? ~u : (u | 0x80000000u);
    bu = (uint16_t)(u >> 20);                      // top 12 bits -> bucket
    atomicAdd(&hist[bu], 1);
  }
  candB[idx] = bu;                                 // snapshot for k_apply
}

__global__ void k_select(int* scal, const int* hist) {
  if (threadIdx.x != 0 || scal[6]) return;
  long long total = 0;
  for (int b = 0; b < NBUCK; ++b) total += hist[b];
  if (!total) { scal[6] = 1; return; }
  long long bk = scal[3], cum = 0; int T = 0;
  for (int b = 0; b < NBUCK; ++b) {
    if (cum + hist[b] <= bk) { cum += hist[b]; T = b + 1; } else break;
  }
  if (cum == 0)                                    // never stall: take the
    for (int b = 0; b < NBUCK; ++b)                // smallest-cost bucket
      if (hist[b]) { T = b + 1; break; }
  scal[4] = T;
}

__global__ void k_apply(uint8_t* alive, const uint16_t* __restrict__ candB,
                        int* scal) {
  if (scal[6]) return;
  int idx = blockIdx.x * blockDim.x + threadIdx.x;
  if (idx >= CAND_CELLS) return;
  int bu = candB[idx];
  if (bu >= scal[4]) return;
  int l, rem, n; d_locate(idx, l, rem, n);
  int r = rem / n, c = rem % n, cn = n << 1;
  alive[d_cellOff(l) + rem] = 1;
  uint8_t* ch = alive + d_cellOff(l - 1);
  int b = (2 * r) * cn + 2 * c;
  ch[b] = 0; ch[b + 1] = 0; ch[b + cn] = 0; ch[b + cn + 1] = 0;
  atomicAdd(&scal[5], 1);
}

__global__ void k_finish_pass(int* scal) {
  scal[0] -= 3 * scal[5];
  scal[5] = 0;
}

// ---------------------------------------------------------------------------
// Phase 4: outputs
// ---------------------------------------------------------------------------
__global__ void k_copy_err(const float* __restrict__ errors,
                           float* __restrict__ out) {
  int idx = blockIdx.x * blockDim.x + threadIdx.x;
  if (idx < TOTAL_CELLS) out[idx] = errors[idx];
}

__global__ void k_lm(const uint8_t* __restrict__ alive, int* __restrict__ lm) {
  int idx = blockIdx.x * blockDim.x + threadIdx.x;
  if (idx >= NB0 * NB0) return;
  int r = idx / NB0, c = idx % NB0, v = -1;
  for (int l = 0; l < NLEV; ++l) {
    int n = NB0 >> l;
    if (alive[d_cellOff(l) + (size_t)(r >> l) * n + (c >> l)]) v = l;
  }
  lm[idx] = v;   // int32 bit-exact in the float-typed tail of d_out
}

// ---------------------------------------------------------------------------
extern "C" void kernel_launch(void* const* d_in, const int* in_sizes, int n_in,
                              void* d_out, int out_size, void* d_ws,
                              size_t ws_size, hipStream_t stream) {
  (void)in_sizes; (void)n_in; (void)out_size; (void)ws_size;
  const float* img      = (const float*)d_in[0];
  const int*   fixedLen = (const int*)d_in[1];

  // Workspace carve-up (~41.3 MB total)
  char* w = (char*)d_ws;
  size_t statsBytes = (size_t)CH * TOTAL_CELLS * sizeof(float2);   // 33.6 MB
  size_t errBytes   = (size_t)TOTAL_CELLS * sizeof(float);         //  5.6 MB
  size_t alvBytes   = (size_t)TOTAL_CELLS;                         //  1.4 MB
  size_t candBytes  = (size_t)CAND_CELLS * sizeof(uint16_t);       //  0.7 MB
  float2*   stats  = (float2*)w;
  float*    errors = (float*)(w + statsBytes);
  uint8_t*  alive  = (uint8_t*)(w + statsBytes + errBytes);
  uint16_t* candB  = (uint16_t*)(w + statsBytes + errBytes +
                                 ((alvBytes + 1) & ~(size_t)1));
  int* scal = (int*)(w + ((statsBytes + errBytes + alvBytes + candBytes + 255)
                          & ~(size_t)255));
  int* hist = scal + 16;

  const int thr = 256;
  k_init<<<(TOTAL_CELLS + thr - 1) / thr, thr, 0, stream>>>(alive, scal,
                                                            fixedLen);
  k_stats0<<<(CH * 256 * 256) / 8, thr, 0, stream>>>(img, stats);
  for (int l = 0; l < NLEV - 1; ++l) {
    int np = NB0 >> (l + 1);
    k_agg<<<(np * np + thr - 1) / thr, thr, 0, stream>>>(stats, l);
  }
  for (int l = 0; l < NLEV; ++l) {
    int n = NB0 >> l;
    k_err<<<(n * n + thr - 1) / thr, thr, 0, stream>>>(stats, errors, l);
  }
  for (int p = 0; p < NPASS; ++p) {
    k_pass_setup<<<1, 256, 0, stream>>>(scal, hist);
    k_candidates<<<(CAND_CELLS + thr - 1) / thr, thr, 0, stream>>>(
        errors, alive, candB, scal, hist);
    k_select<<<1, 1, 0, stream>>>(scal, hist);
    k_apply<<<(CAND_CELLS + thr - 1) / thr, thr, 0, stream>>>(alive, candB,
                                                              scal);
    k_finish_pass<<<1, 1, 0, stream>>>(scal);
  }
  k_copy_err<<<(TOTAL_CELLS + thr - 1) / thr, thr, 0, stream>>>(
      errors, (float*)d_out);
  k_lm<<<(NB0 * NB0 + thr - 1) / thr, thr, 0, stream>>>(
      alive, (int*)d_out + TOTAL_CELLS);
}